// LSTM_12962211299467
// MI455X (gfx1250) — compile-verified
//
#include <hip/hip_runtime.h>

typedef __attribute__((ext_vector_type(2))) float v2f;
typedef __attribute__((ext_vector_type(8))) float v8f;

#define T_STEPS 9
#define I_DIM   57
#define H_DIM   2
#define G_DIM   8     // 4*H gate rows
#define KCH     14    // full 4-wide K chunks (k = 0..55); k = 56 handled as VALU fixup

__device__ __forceinline__ float fast_sigmoid(float x) {
    return 1.0f / (1.0f + __expf(-x));
}
__device__ __forceinline__ float fast_tanh(float x) {
    return 2.0f / (1.0f + __expf(-2.0f * x)) - 1.0f;
}

// One wave32 processes 16 batch rows end-to-end (all T timesteps).
// A = W_ih (gates as M rows, padded 8->16), B = x_t (batches as N columns),
// so D lane n holds all 8 gate pre-activations of batch n in 8 VGPRs.
__global__ __launch_bounds__(256) void lstm_wmma_kernel(
    const float* __restrict__ x,
    const float* __restrict__ W_ih,
    const float* __restrict__ W_hh,
    const float* __restrict__ b_ih,
    const float* __restrict__ b_hh,
    const float* __restrict__ fc_w,
    const float* __restrict__ fc_b,
    float* __restrict__ out,
    int B)
{
    const int lane = threadIdx.x & 31;
    const int wid  = threadIdx.x >> 5;
    const int gw   = blockIdx.x * (blockDim.x >> 5) + wid;
    const int b_base = gw * 16;
    if (b_base >= B) return;                 // wave-uniform exit

    const int m  = lane & 15;                // A: gate row (m<8 valid); B/D: batch column
    const int kh = lane >> 4;                // K-half selector within a 4-wide chunk

    // ---- small uniform weights -> scalar loads ----
    float bias[G_DIM], w56[G_DIM], whh0[G_DIM], whh1[G_DIM];
    #pragma unroll
    for (int r = 0; r < G_DIM; ++r) {
        bias[r] = b_ih[r] + b_hh[r];
        w56[r]  = W_ih[r * I_DIM + 56];
        whh0[r] = W_hh[r * H_DIM + 0];
        whh1[r] = W_hh[r * H_DIM + 1];
    }
    const float fcw0 = fc_w[0], fcw1 = fc_w[1], fcb = fc_b[0];

    // ---- A fragments: W_ih rows (zero-padded to 16), K chunks of 4 ----
    // 32-bit A 16x4 layout: lane (m, kh) holds A[m][4*kc + 2*kh + {0,1}]
    v2f aw[KCH];
    #pragma unroll
    for (int kc = 0; kc < KCH; ++kc) {
        const int kb = kc * 4 + kh * 2;
        float a0 = 0.0f, a1 = 0.0f;
        if (m < G_DIM) {
            a0 = W_ih[m * I_DIM + kb];
            a1 = W_ih[m * I_DIM + kb + 1];
        }
        aw[kc].x = a0;
        aw[kc].y = a1;
    }

    const int brow = (b_base + m < B) ? (b_base + m) : (B - 1);
    const float* __restrict__ xrow = x + (size_t)brow * (T_STEPS * I_DIM);

    float h0 = 0.f, h1 = 0.f, c0 = 0.f, c1 = 0.f;

    for (int t = 0; t < T_STEPS; ++t) {
        const float* __restrict__ xt = xrow + t * I_DIM;

        v8f acc;
        #pragma unroll
        for (int r = 0; r < G_DIM; ++r) acc[r] = bias[r];   // C init = b_ih + b_hh

        // gx = W_ih @ x_t  over k = 0..55 via 14 f32 WMMAs
        #pragma unroll
        for (int kc = 0; kc < KCH; ++kc) {
            const int kb = kc * 4 + kh * 2;
            v2f bfrag;                        // B 4x16: lane (n=m, kh) holds B[2*kh+{0,1}][n]
            bfrag.x = xt[kb];
            bfrag.y = xt[kb + 1];
            acc = __builtin_amdgcn_wmma_f32_16x16x4_f32(
                /*neg_a=*/false, aw[kc],
                /*neg_b=*/false, bfrag,
                /*c_mod=*/(short)0, acc,
                /*reuse_a=*/false, /*reuse_b=*/false);
        }

        // k = 56 remainder + recurrent term, then LSTM cell (all per-lane scalar)
        const float x56 = xt[56];
        float gp[G_DIM];
        #pragma unroll
        for (int r = 0; r < G_DIM; ++r)
            gp[r] = acc[r] + w56[r] * x56 + whh0[r] * h0 + whh1[r] * h1;

        const float i0 = fast_sigmoid(gp[0]), i1 = fast_sigmoid(gp[1]);
        const float f0 = fast_sigmoid(gp[2]), f1 = fast_sigmoid(gp[3]);
        const float g0 = fast_tanh(gp[4]),    g1 = fast_tanh(gp[5]);
        const float o0 = fast_sigmoid(gp[6]), o1 = fast_sigmoid(gp[7]);
        c0 = f0 * c0 + i0 * g0;
        c1 = f1 * c1 + i1 * g1;
        h0 = o0 * fast_tanh(c0);
        h1 = o1 * fast_tanh(c1);
    }

    if (lane < 16 && (b_base + m) < B)
        out[b_base + m] = h0 * fcw0 + h1 * fcw1 + fcb;
}

extern "C" void kernel_launch(void* const* d_in, const int* in_sizes, int n_in,
                              void* d_out, int out_size, void* d_ws, size_t ws_size,
                              hipStream_t stream) {
    (void)n_in; (void)d_ws; (void)ws_size; (void)out_size;
    const float* x    = (const float*)d_in[0];
    const float* W_ih = (const float*)d_in[1];
    const float* W_hh = (const float*)d_in[2];
    const float* b_ih = (const float*)d_in[3];
    const float* b_hh = (const float*)d_in[4];
    const float* fc_w = (const float*)d_in[5];
    const float* fc_b = (const float*)d_in[6];
    float* out = (float*)d_out;

    const int B = in_sizes[0] / (T_STEPS * I_DIM);   // 65536
    const int batches_per_block = 8 * 16;            // 8 waves * 16 rows
    const int grid = (B + batches_per_block - 1) / batches_per_block;

    lstm_wmma_kernel<<<grid, 256, 0, stream>>>(x, W_ih, W_hh, b_ih, b_hh,
                                               fc_w, fc_b, out, B);
}